// Fbank_80762565034284
// MI455X (gfx1250) — compile-verified
//
#include <hip/hip_runtime.h>
#include <math.h>

// Problem constants (match reference)
#define NFFT      512
#define HOP       256
#define NMELS     64
#define NFREQ     257          // NFFT/2+1
#define BATCH     32
#define WAVE_LEN  960000
#define NFRAMES   3751         // (960000 + 512 - 512)/256 + 1
#define M_ROWS    (BATCH*NFRAMES)   // 120032, divisible by 32
#define CHUNK     256
#define NCHUNK    15           // ceil(3751/256)

typedef __attribute__((ext_vector_type(2))) float v2f;
typedef __attribute__((ext_vector_type(8))) float v8f;

// ---------------------------------------------------------------------------
// Kernel 1: framed + Hann-windowed 512-pt FFT magnitude (one block per frame)
// Radix-2 DIT in LDS, bit-reversed input, 9 stages, 256 butterflies/stage.
// ---------------------------------------------------------------------------
__global__ __launch_bounds__(256)
void stft_mag_kernel(const float* __restrict__ wav, float* __restrict__ mag) {
  __shared__ float re[NFFT];
  __shared__ float im[NFFT];
  const int fidx = blockIdx.x;            // b*NFRAMES + t
  const int b = fidx / NFRAMES;
  const int t = fidx - b * NFRAMES;
  const float* x = wav + (size_t)b * WAVE_LEN;
  const int tid = threadIdx.x;

  // load 2 samples/thread with reflect padding + Hann window, bit-reversed store
  #pragma unroll
  for (int i = 0; i < 2; ++i) {
    int n = tid + i * 256;
    int p = t * HOP + n;                  // index into padded signal
    int s = p - (NFFT / 2);               // reflect pad of 256
    if (s < 0) s = -s;
    if (s >= WAVE_LEN) s = 2 * (WAVE_LEN - 1) - s;
    float w = 0.5f - 0.5f * __cosf(6.283185307179586f * (float)n / (float)NFFT);
    int r = (int)(__brev((unsigned)n) >> 23);   // 9-bit reverse
    re[r] = w * x[s];
    im[r] = 0.f;
  }
  __syncthreads();

  // 9 radix-2 DIT stages; butterfly pairs are disjoint per stage
  for (int stage = 0; stage < 9; ++stage) {
    int half = 1 << stage;
    int grp  = tid >> stage;
    int pos  = tid & (half - 1);
    int i0 = (grp << (stage + 1)) + pos;
    int i1 = i0 + half;
    float ang = -6.283185307179586f * (float)pos / (float)(2 * half);
    float sw, cw;
    __sincosf(ang, &sw, &cw);
    float xr = re[i1], xi = im[i1];
    float tr = cw * xr - sw * xi;
    float ti = cw * xi + sw * xr;
    float ur = re[i0], ui = im[i0];
    re[i0] = ur + tr;  im[i0] = ui + ti;
    re[i1] = ur - tr;  im[i1] = ui - ti;
    __syncthreads();
  }

  for (int k = tid; k < NFREQ; k += 256) {
    float vr = re[k], vi = im[k];
    mag[(size_t)fidx * NFREQ + k] = sqrtf(vr * vr + vi * vi);
  }
}

// ---------------------------------------------------------------------------
// EWMA scan, 3-phase chunked parallelization.
// c_t = a*c_{t-1} + (1-a)*m_t ; carry init = mag[:,0,:]  (so avg[0]==mag[0])
// ---------------------------------------------------------------------------

// Phase A: per (b, chunk, k) homogeneous partial (carry-in = 0)
__global__ __launch_bounds__(288)
void ewma_chunk_partial(const float* __restrict__ mag, const float* __restrict__ alpha,
                        float* __restrict__ pend) {
  int bc = blockIdx.x;                  // b*NCHUNK + ch
  int b  = bc / NCHUNK;
  int ch = bc - b * NCHUNK;
  int k  = threadIdx.x;
  if (k >= NFREQ) return;
  float a  = 1.f / (1.f + __expf(-alpha[k]));
  float om = 1.f - a;
  int t0 = ch * CHUNK;
  int t1 = min(t0 + CHUNK, NFRAMES);
  const float* mrow = mag + (size_t)b * NFRAMES * NFREQ + k;
  float c = 0.f;
  for (int t = t0; t < t1; ++t)
    c = a * c + om * mrow[(size_t)t * NFREQ];
  pend[((size_t)ch * BATCH + b) * NFREQ + k] = c;
}

// Phase B: sequential combine of NCHUNK chunk carries per (b, k)
__global__ __launch_bounds__(288)
void ewma_chunk_scan(const float* __restrict__ mag, const float* __restrict__ alpha,
                     const float* __restrict__ pend, float* __restrict__ cinit) {
  int b = blockIdx.x;
  int k = threadIdx.x;
  if (k >= NFREQ) return;
  float a = 1.f / (1.f + __expf(-alpha[k]));
  float c = mag[(size_t)b * NFRAMES * NFREQ + k];     // initial carry = mag[:,0,:]
  for (int ch = 0; ch < NCHUNK; ++ch) {
    cinit[((size_t)ch * BATCH + b) * NFREQ + k] = c;
    int L = min(CHUNK, NFRAMES - ch * CHUNK);
    c = __powf(a, (float)L) * c + pend[((size_t)ch * BATCH + b) * NFREQ + k];
  }
}

// Phase C: replay chunk with exact carry, write norm_mag in place over mag
__global__ __launch_bounds__(288)
void ewma_normalize(float* __restrict__ mag, const float* __restrict__ alpha,
                    const float* __restrict__ cinit) {
  int bc = blockIdx.x;
  int b  = bc / NCHUNK;
  int ch = bc - b * NCHUNK;
  int k  = threadIdx.x;
  if (k >= NFREQ) return;
  float a  = 1.f / (1.f + __expf(-alpha[k]));
  float om = 1.f - a;
  float c = cinit[((size_t)ch * BATCH + b) * NFREQ + k];
  int t0 = ch * CHUNK;
  int t1 = min(t0 + CHUNK, NFRAMES);
  float* mrow = mag + (size_t)b * NFRAMES * NFREQ + k;
  for (int t = t0; t < t1; ++t) {
    float m = mrow[(size_t)t * NFREQ];
    c = a * c + om * m;
    mrow[(size_t)t * NFREQ] = m / (c + 1e-8f);
  }
}

// ---------------------------------------------------------------------------
// Kernel 3: mel projection via V_WMMA_F32_16X16X4_F32.
// (M_ROWS x 257) * (257 x 64). One 16x16 output tile per wave; 8 waves/block
// cover a 32x64 slab. K consumed 4 at a time (65 steps, zero-padded tail).
// No divergence: EXEC stays all-ones as WMMA requires.
// ---------------------------------------------------------------------------
__global__ __launch_bounds__(256)
void mel_matmul_wmma(const float* __restrict__ normmag, const float* __restrict__ mel,
                     float* __restrict__ out) {
  const int wave = threadIdx.x >> 5;
  const int lane = threadIdx.x & 31;
  const int half = lane >> 4;            // 0: lanes 0-15, 1: lanes 16-31
  const int lm   = lane & 15;
  const int m_tile = blockIdx.x * 2 + (wave >> 2);
  const int n0     = (wave & 3) * 16;
  const int row    = m_tile * 16 + lm;   // A row for this lane

  const float* arow = normmag + (size_t)row * NFREQ;
  v8f acc = {};
  for (int kk = 0; kk < 65; ++kk) {
    int k0 = kk * 4 + half * 2;          // lanes 16-31 handle K+2/K+3
    v2f a, bf;
    a.x  = (k0     < NFREQ) ? arow[k0]     : 0.f;
    a.y  = (k0 + 1 < NFREQ) ? arow[k0 + 1] : 0.f;
    bf.x = (k0     < NFREQ) ? mel[(size_t)k0       * NMELS + n0 + lm] : 0.f;
    bf.y = (k0 + 1 < NFREQ) ? mel[(size_t)(k0 + 1) * NMELS + n0 + lm] : 0.f;
    // 8 args: (neg_a, A, neg_b, B, c_mod, C, reuse_a, reuse_b)
    acc = __builtin_amdgcn_wmma_f32_16x16x4_f32(
        false, a, false, bf, (short)0, acc, false, false);
  }

  // C/D layout: VGPR r -> row (m_tile*16 + r + 8*half), col (n0 + lm)
  float* orow = out + ((size_t)m_tile * 16 + 8 * half) * NMELS + n0 + lm;
  #pragma unroll
  for (int r = 0; r < 8; ++r)
    orow[(size_t)r * NMELS] = acc[r] + 1e-6f;
}

// ---------------------------------------------------------------------------
extern "C" void kernel_launch(void* const* d_in, const int* in_sizes, int n_in,
                              void* d_out, int out_size, void* d_ws, size_t ws_size,
                              hipStream_t stream) {
  const float* wav   = (const float*)d_in[0];   // (32, 960000) f32
  const float* alpha = (const float*)d_in[1];   // (1, 257)     f32
  const float* mel   = (const float*)d_in[2];   // (257, 64)    f32
  float* out = (float*)d_out;                   // (32, 3751, 64) f32

  // workspace layout
  float* mag   = (float*)d_ws;                                  // M_ROWS*NFREQ
  size_t NMAG  = (size_t)M_ROWS * NFREQ;
  float* pend  = mag + NMAG;                                    // NCHUNK*B*NFREQ
  float* cinit = pend + (size_t)NCHUNK * BATCH * NFREQ;

  stft_mag_kernel   <<<M_ROWS,        256, 0, stream>>>(wav, mag);
  ewma_chunk_partial<<<BATCH*NCHUNK,  288, 0, stream>>>(mag, alpha, pend);
  ewma_chunk_scan   <<<BATCH,         288, 0, stream>>>(mag, alpha, pend, cinit);
  ewma_normalize    <<<BATCH*NCHUNK,  288, 0, stream>>>(mag, alpha, cinit);
  mel_matmul_wmma   <<<M_ROWS/32,     256, 0, stream>>>(mag, mel, out);
}